// SortGAT_25065429139615
// MI455X (gfx1250) — compile-verified
//
#include <hip/hip_runtime.h>
#include <hip/hip_bf16.h>

// ---------------------------------------------------------------------------
// GATConv forward for MI455X (gfx1250, wave32).
//   xp = x @ W                      -> fp32 WMMA 16x16x4 GEMM
//   a_src/a_dst = <xp, att>         -> wave-per-(node,head) reduction
//   scatter softmax over dst edges  -> atomicMax(u32 ordered key) + atomicAdd
//   out = mean_h segment_sum(alpha * xp[src]) + bias
// ---------------------------------------------------------------------------

#define GAT_N      50000
#define GAT_E      800000
#define GAT_E2     (GAT_E + GAT_N)   // edges + self loops
#define GAT_DIN    128
#define GAT_H      4
#define GAT_D      64
#define GAT_CH     (GAT_H * GAT_D)   // 256
#define GAT_MT     (GAT_N / 16)      // 3125 M-tiles (N divisible by 16)

typedef float v2f __attribute__((ext_vector_type(2)));
typedef float v8f __attribute__((ext_vector_type(8)));

// Monotonic float <-> uint key for atomicMax-based segment max.
__device__ __forceinline__ unsigned gat_fkey(float f) {
  unsigned u = __float_as_uint(f);
  return u ^ ((unsigned)((int)u >> 31) | 0x80000000u);
}
__device__ __forceinline__ float gat_fdec(unsigned k) {
  unsigned u = (k & 0x80000000u) ? (k ^ 0x80000000u) : ~k;
  return __uint_as_float(u);
}

__device__ __forceinline__ float gat_leaky(float x) {
  return x >= 0.f ? x : 0.2f * x;
}

// ---------------------------------------------------------------------------
// 0) Init: out = bias (so no epilogue bias pass), maxkey = 0 (< key(-inf)),
//    sums = 0.
// ---------------------------------------------------------------------------
__global__ void gat_init(float* __restrict__ out, const float* __restrict__ bias,
                         unsigned* __restrict__ maxk, float* __restrict__ sumv) {
  int i = blockIdx.x * blockDim.x + threadIdx.x;
  if (i < GAT_N * GAT_D) out[i] = bias[i & (GAT_D - 1)];
  if (i < GAT_N * GAT_H) { maxk[i] = 0u; sumv[i] = 0.f; }
}

// ---------------------------------------------------------------------------
// 1) xp = x @ W with V_WMMA_F32_16X16X4_F32.
//    Block = 128 threads (4 waves); each wave owns one 16x16 output tile,
//    4 M-tiles per block, grid.y covers the 16 N-tiles of the 256-wide output.
//    A (16x4): lane 0-15 -> K=k,k+1 ; lane 16-31 -> K=k+2,k+3 (per ISA layout)
//    B (4x16): mirrored K striping; C/D: VGPR v -> M = v + 8*(lane/16).
// ---------------------------------------------------------------------------
__global__ __launch_bounds__(128) void gat_gemm_wmma(const float* __restrict__ x,
                                                     const float* __restrict__ W,
                                                     float* __restrict__ xp) {
  const int lane  = threadIdx.x & 31;
  const int wave  = threadIdx.x >> 5;
  const int mTile = blockIdx.x * 4 + wave;
  if (mTile >= GAT_MT) return;                    // uniform per wave

  const int m0    = mTile * 16;
  const int n0    = blockIdx.y * 16;
  const int l16   = lane & 15;
  const int khalf = (lane >> 4) * 2;              // 0 for lanes 0-15, 2 for 16-31
  const int row   = m0 + l16;                     // A-matrix row for this lane
  const int col   = n0 + l16;                     // B/C column for this lane

  const float* __restrict__ xrow = x + (size_t)row * GAT_DIN;

  v8f acc = {};
#pragma unroll
  for (int kk = 0; kk < GAT_DIN; kk += 4) {
    const int klo = kk + khalf;
    v2f a = *(const v2f*)(xrow + klo);            // A[row, klo], A[row, klo+1]
    v2f b;
    b.x = W[(size_t)klo * GAT_CH + col];          // B[klo,   col]
    b.y = W[(size_t)(klo + 1) * GAT_CH + col];    // B[klo+1, col]
    acc = __builtin_amdgcn_wmma_f32_16x16x4_f32(
        /*neg_a=*/false, a, /*neg_b=*/false, b,
        /*c_mod=*/(short)0, acc, /*reuse_a=*/false, /*reuse_b=*/false);
  }

#pragma unroll
  for (int v = 0; v < 8; ++v) {
    const int r = m0 + v + 8 * (lane >> 4);
    xp[(size_t)r * GAT_CH + col] = acc[v];
  }
}

// ---------------------------------------------------------------------------
// 2) Per-node logits: a_src[n,h] = sum_d xp[n,h,d]*att_src[h,d] (same for dst).
//    One wave32 per (node, head); 2 elems/lane + shfl_xor tree reduce.
// ---------------------------------------------------------------------------
__global__ __launch_bounds__(256) void gat_logits(const float* __restrict__ xp,
                                                  const float* __restrict__ att_src,
                                                  const float* __restrict__ att_dst,
                                                  float* __restrict__ a_src,
                                                  float* __restrict__ a_dst) {
  const int w    = blockIdx.x * 8 + (threadIdx.x >> 5);
  const int lane = threadIdx.x & 31;
  if (w >= GAT_N * GAT_H) return;
  const int n = w >> 2;
  const int h = w & 3;

  const float* __restrict__ px = xp + (size_t)n * GAT_CH + h * GAT_D;
  const float* __restrict__ ps = att_src + h * GAT_D;
  const float* __restrict__ pd = att_dst + h * GAT_D;

  const float x0 = px[lane], x1 = px[lane + 32];
  float as = x0 * ps[lane] + x1 * ps[lane + 32];
  float ad = x0 * pd[lane] + x1 * pd[lane + 32];
#pragma unroll
  for (int off = 16; off > 0; off >>= 1) {
    as += __shfl_xor(as, off, 32);
    ad += __shfl_xor(ad, off, 32);
  }
  if (lane == 0) {
    a_src[n * GAT_H + h] = as;
    a_dst[n * GAT_H + h] = ad;
  }
}

// ---------------------------------------------------------------------------
// 3) Segment max over incoming edges: one thread per (edge, head).
// ---------------------------------------------------------------------------
__global__ __launch_bounds__(256) void gat_edge_max(const int* __restrict__ ei,
                                                    const float* __restrict__ a_src,
                                                    const float* __restrict__ a_dst,
                                                    unsigned* __restrict__ maxk) {
  const int t = blockIdx.x * blockDim.x + threadIdx.x;
  if (t >= GAT_E2 * GAT_H) return;
  const int e = t >> 2;
  const int h = t & 3;
  int s, d;
  if (e < GAT_E) { s = ei[e]; d = ei[GAT_E + e]; }
  else           { s = d = e - GAT_E; }
  const float sc = gat_leaky(a_src[s * GAT_H + h] + a_dst[d * GAT_H + h]);
  atomicMax(&maxk[d * GAT_H + h], gat_fkey(sc));
}

// ---------------------------------------------------------------------------
// 4) Segment sum of exp(score - max): one thread per (edge, head).
// ---------------------------------------------------------------------------
__global__ __launch_bounds__(256) void gat_edge_sum(const int* __restrict__ ei,
                                                    const float* __restrict__ a_src,
                                                    const float* __restrict__ a_dst,
                                                    const unsigned* __restrict__ maxk,
                                                    float* __restrict__ sumv) {
  const int t = blockIdx.x * blockDim.x + threadIdx.x;
  if (t >= GAT_E2 * GAT_H) return;
  const int e = t >> 2;
  const int h = t & 3;
  int s, d;
  if (e < GAT_E) { s = ei[e]; d = ei[GAT_E + e]; }
  else           { s = d = e - GAT_E; }
  const float sc = gat_leaky(a_src[s * GAT_H + h] + a_dst[d * GAT_H + h]);
  const float w  = __expf(sc - gat_fdec(maxk[d * GAT_H + h]));
  atomicAdd(&sumv[d * GAT_H + h], w);
}

// ---------------------------------------------------------------------------
// 5) Aggregation fused with head-mean:
//    out[dst, dd] += (1/H) * sum_h alpha_h(e) * xp[src, h, dd]
//    64 threads per edge (dd dimension), 4 edges per 256-thread block.
//    Alpha recomputed per thread (scalar loads broadcast from L2/WGP$).
// ---------------------------------------------------------------------------
__global__ __launch_bounds__(256) void gat_agg(const int* __restrict__ ei,
                                               const float* __restrict__ a_src,
                                               const float* __restrict__ a_dst,
                                               const unsigned* __restrict__ maxk,
                                               const float* __restrict__ sumv,
                                               const float* __restrict__ xp,
                                               float* __restrict__ out) {
  const int e  = blockIdx.x * 4 + (threadIdx.x >> 6);
  const int dd = threadIdx.x & 63;
  if (e >= GAT_E2) return;
  int s, d;
  if (e < GAT_E) { s = ei[e]; d = ei[GAT_E + e]; }
  else           { s = d = e - GAT_E; }

  const float* __restrict__ xs = xp + (size_t)s * GAT_CH;
  float acc = 0.f;
#pragma unroll
  for (int h = 0; h < GAT_H; ++h) {
    const float sc = gat_leaky(a_src[s * GAT_H + h] + a_dst[d * GAT_H + h]);
    const float al = __expf(sc - gat_fdec(maxk[d * GAT_H + h])) /
                     (sumv[d * GAT_H + h] + 1e-16f);
    acc += al * xs[h * GAT_D + dd];
  }
  atomicAdd(&out[(size_t)d * GAT_D + dd], 0.25f * acc);
}

// ---------------------------------------------------------------------------
// Launch
// ---------------------------------------------------------------------------
extern "C" void kernel_launch(void* const* d_in, const int* in_sizes, int n_in,
                              void* d_out, int out_size, void* d_ws, size_t ws_size,
                              hipStream_t stream) {
  const float* x        = (const float*)d_in[0];
  const int*   ei       = (const int*)d_in[1];   // int32 (JAX x64 disabled)
  const float* W        = (const float*)d_in[2];
  const float* att_src  = (const float*)d_in[3];
  const float* att_dst  = (const float*)d_in[4];
  const float* bias     = (const float*)d_in[5];
  float*       out      = (float*)d_out;

  // Workspace layout (floats):
  //   [0,            12,800,000) xp   = N*256
  //   [12,800,000,   13,000,000) a_src = N*4
  //   [13,000,000,   13,200,000) a_dst = N*4
  //   [13,200,000,   13,400,000) maxkey (unsigned) = N*4
  //   [13,400,000,   13,600,000) sumv  = N*4          (total 54.4 MB)
  float*    ws    = (float*)d_ws;
  float*    xp    = ws;
  float*    a_src = ws + (size_t)GAT_N * GAT_CH;
  float*    a_dst = a_src + (size_t)GAT_N * GAT_H;
  unsigned* maxk  = (unsigned*)(a_dst + (size_t)GAT_N * GAT_H);
  float*    sumv  = (float*)(maxk + (size_t)GAT_N * GAT_H);

  // 0) init out=bias, maxkey=0, sum=0
  {
    int total = GAT_N * GAT_D;
    gat_init<<<(total + 255) / 256, 256, 0, stream>>>(out, bias, maxk, sumv);
  }
  // 1) projection GEMM (WMMA f32 16x16x4)
  {
    dim3 grid((GAT_MT + 3) / 4, GAT_CH / 16);
    gat_gemm_wmma<<<grid, 128, 0, stream>>>(x, W, xp);
  }
  // 2) per-node attention logits
  {
    int waves = GAT_N * GAT_H;
    gat_logits<<<(waves + 7) / 8, 256, 0, stream>>>(xp, att_src, att_dst, a_src, a_dst);
  }
  // 3) scatter max
  {
    int total = GAT_E2 * GAT_H;
    gat_edge_max<<<(total + 255) / 256, 256, 0, stream>>>(ei, a_src, a_dst, maxk);
  }
  // 4) scatter exp-sum
  {
    int total = GAT_E2 * GAT_H;
    gat_edge_sum<<<(total + 255) / 256, 256, 0, stream>>>(ei, a_src, a_dst, maxk, sumv);
  }
  // 5) weighted aggregation + head mean (bias already in out)
  {
    gat_agg<<<(GAT_E2 + 3) / 4, 256, 0, stream>>>(ei, a_src, a_dst, maxk, sumv, xp, out);
  }
}